// GNNEncoder_66640712565461
// MI455X (gfx1250) — compile-verified
//
#include <hip/hip_runtime.h>
#include <hip/hip_bf16.h>
#include <math.h>

// Problem constants (match reference)
#define NN   50000
#define EE   800000
#define DIN  384
#define DH   256
#define ED   32
#define NH   4      // heads
#define NC   64     // channels per head
#define ETOT (EE + NN)

typedef float v2f __attribute__((ext_vector_type(2)));
typedef float v8f __attribute__((ext_vector_type(8)));

// ---------- helpers ----------
__device__ __forceinline__ void atomAddF(float* p, float v) {
  __hip_atomic_fetch_add(p, v, __ATOMIC_RELAXED, __HIP_MEMORY_SCOPE_AGENT);
}
__device__ __forceinline__ void atomAddFShared(float* p, float v) {
  __hip_atomic_fetch_add(p, v, __ATOMIC_RELAXED, __HIP_MEMORY_SCOPE_WORKGROUP);
}
// monotonic float <-> uint encoding so atomicMax(u32) implements float max
__device__ __forceinline__ unsigned fenc(float f) {
  unsigned u = __float_as_uint(f);
  return (u & 0x80000000u) ? ~u : (u | 0x80000000u);
}
__device__ __forceinline__ float fdec(unsigned u) {
  return (u & 0x80000000u) ? __uint_as_float(u & 0x7FFFFFFFu) : __uint_as_float(~u);
}

// ---------- WMMA f32 GEMM:  out[M,Nw] = act(A[M,K] @ W[K,Nw] + bias) ----------
// Each wave: 16 rows x 64 cols (4 WMMA accumulators, A-frag reused x4).
// Block: 8 waves -> 128 rows. grid = (ceil(M/128), Nw/64).
// Out-of-range rows are CLAMPED (not predicated): they compute garbage that is
// never stored, keeping the inner loop branch-free with full-EXEC WMMAs.
template <int ACT>  // 0 = none, 1 = relu
__global__ __launch_bounds__(256)
void gemm_bias_act(const float* __restrict__ A, const float* __restrict__ W,
                   const float* __restrict__ bias, float* __restrict__ out,
                   int M, int K, int Nw) {
  const int lane = threadIdx.x & 31;
  const int wave = threadIdx.x >> 5;
  const int row0 = blockIdx.x * 128 + wave * 16;
  const int col0 = blockIdx.y * 64;
  const int half = lane >> 4;    // 0: K=0,1 ; 1: K=2,3
  const int l15  = lane & 15;

  const int arow = min(row0 + l15, M - 1);          // clamp: branch-free loads
  const float* __restrict__ Arow = A + (size_t)arow * K;

  v8f acc[4];
#pragma unroll
  for (int t = 0; t < 4; ++t) {
    float b = bias[col0 + t * 16 + l15];
#pragma unroll
    for (int i = 0; i < 8; ++i) acc[t][i] = b;
  }

  for (int k = 0; k < K; k += 4) {
    const int ka = k + half * 2;
    const v2f a = *(const v2f*)(Arow + ka);         // global_load_b64
#pragma unroll
    for (int t = 0; t < 4; ++t) {
      const int col = col0 + t * 16 + l15;
      v2f b;
      b.x = W[(size_t)ka * Nw + col];
      b.y = W[(size_t)(ka + 1) * Nw + col];
      acc[t] = __builtin_amdgcn_wmma_f32_16x16x4_f32(
          false, a, false, b, (short)0, acc[t], false, false);
    }
  }

#pragma unroll
  for (int t = 0; t < 4; ++t) {
    const int col = col0 + t * 16 + l15;
#pragma unroll
    for (int i = 0; i < 8; ++i) {
      const int r = row0 + i + half * 8;  // C/D layout: VGPR i -> M=i (lanes<16) / M=i+8
      if (r < M) {
        float v = acc[t][i];
        if (ACT == 1) v = v > 0.0f ? v : 0.0f;
        out[(size_t)r * Nw + col] = v;
      }
    }
  }
}

// ---------- edge_attr mean (for self-loop fill_value='mean') ----------
__global__ __launch_bounds__(256)
void ea_sum_kernel(const float* __restrict__ ea, float* __restrict__ easum) {
  __shared__ float s[ED];
  if (threadIdx.x < ED) s[threadIdx.x] = 0.0f;
  __syncthreads();
  const size_t total  = (size_t)EE * ED;
  const size_t stride = (size_t)gridDim.x * blockDim.x;  // multiple of 32
  size_t idx = (size_t)blockIdx.x * blockDim.x + threadIdx.x;
  const int dim = (int)(idx & (ED - 1));  // constant per thread (stride % 32 == 0)
  float local = 0.0f;
  for (; idx < total; idx += stride) local += ea[idx];
  atomAddFShared(&s[dim], local);
  __syncthreads();
  if (threadIdx.x < ED) atomAddF(&easum[threadIdx.x], s[threadIdx.x]);
}

__global__ void ea_scale_kernel(float* easum) {
  if (threadIdx.x < ED) easum[threadIdx.x] *= (1.0f / (float)EE);
}

// ---------- GATv2 pass 1: logits + segment max ----------
// one thread per (edge, head); ee recomputed on the fly (FLOPs are free vs HBM)
__global__ __launch_bounds__(256)
void edge_logits_kernel(const int* __restrict__ srcs, const int* __restrict__ dsts,
                        const float* __restrict__ ea, const float* __restrict__ ea_mean,
                        const float* __restrict__ xl, const float* __restrict__ xr,
                        const float* __restrict__ We, const float* __restrict__ att,
                        float* __restrict__ logits, unsigned* __restrict__ mx) {
  const int idx = blockIdx.x * blockDim.x + threadIdx.x;
  if (idx >= ETOT * NH) return;
  const int e  = idx >> 2;
  const int hh = idx & 3;
  int s, d;
  const float* eav;
  if (e < EE) { s = srcs[e]; d = dsts[e]; eav = ea + (size_t)e * ED; }
  else        { s = d = e - EE;           eav = ea_mean; }

  float ear[ED];
#pragma unroll
  for (int k = 0; k < ED; k += 4) {   // b128 loads (ea rows are 16B aligned)
    const float4 q = *(const float4*)(eav + k);
    ear[k] = q.x; ear[k + 1] = q.y; ear[k + 2] = q.z; ear[k + 3] = q.w;
  }

  const float4* __restrict__ atth = (const float4*)(att + hh * NC);
  const float4* __restrict__ xls  = (const float4*)(xl + (size_t)s * DH + hh * NC);
  const float4* __restrict__ xrd  = (const float4*)(xr + (size_t)d * DH + hh * NC);
  float logit = 0.0f;
  for (int c4 = 0; c4 < NC / 4; ++c4) {
    const float4 l4 = xls[c4];
    const float4 r4 = xrd[c4];
    const float4 a4 = atth[c4];
    const float lv[4] = {l4.x, l4.y, l4.z, l4.w};
    const float rv[4] = {r4.x, r4.y, r4.z, r4.w};
    const float av[4] = {a4.x, a4.y, a4.z, a4.w};
#pragma unroll
    for (int j = 0; j < 4; ++j) {
      const int c = c4 * 4 + j;
      const float* wcol = We + hh * NC + c;  // column of We[32,256]
      float eev = 0.0f;
#pragma unroll
      for (int k = 0; k < ED; ++k) eev = fmaf(ear[k], wcol[(size_t)k * DH], eev);
      float m = lv[j] + rv[j] + eev;
      m = m > 0.0f ? m : 0.2f * m;  // leaky_relu(0.2)
      logit = fmaf(m, av[j], logit);
    }
  }
  logits[idx] = logit;
  atomicMax(mx + (size_t)d * NH + hh, fenc(logit));
}

// ---------- GATv2 pass 2: p = exp(logit - max), segment sum of p ----------
__global__ __launch_bounds__(256)
void edge_p_kernel(const int* __restrict__ dsts, float* __restrict__ logits,
                   const unsigned* __restrict__ mx, float* __restrict__ den) {
  const int idx = blockIdx.x * blockDim.x + threadIdx.x;
  if (idx >= ETOT * NH) return;
  const int e  = idx >> 2;
  const int hh = idx & 3;
  const int d  = (e < EE) ? dsts[e] : (e - EE);
  const float p = __expf(logits[idx] - fdec(mx[(size_t)d * NH + hh]));
  logits[idx] = p;  // reuse buffer for p
  atomAddF(den + (size_t)d * NH + hh, p);
}

// ---------- GATv2 pass 3: acc[dst] += xl[src] * alpha ----------
__global__ __launch_bounds__(256)
void edge_agg_kernel(const int* __restrict__ srcs, const int* __restrict__ dsts,
                     const float* __restrict__ p, const float* __restrict__ den,
                     const float* __restrict__ xl, float* __restrict__ acc) {
  const int idx = blockIdx.x * blockDim.x + threadIdx.x;
  if (idx >= ETOT * NH) return;
  const int e  = idx >> 2;
  const int hh = idx & 3;
  int s, d;
  if (e < EE) { s = srcs[e]; d = dsts[e]; } else { s = d = e - EE; }
  const float alpha = p[idx] / den[(size_t)d * NH + hh];
  const float4* __restrict__ xls = (const float4*)(xl + (size_t)s * DH + hh * NC);
  float* __restrict__ accd = acc + (size_t)d * DH + hh * NC;
#pragma unroll 4
  for (int c4 = 0; c4 < NC / 4; ++c4) {
    const float4 v = xls[c4];                 // b128 gather from L2-resident xl
    atomAddF(accd + c4 * 4 + 0, v.x * alpha); // scatter adds must stay scalar
    atomAddF(accd + c4 * 4 + 1, v.y * alpha);
    atomAddF(accd + c4 * 4 + 2, v.z * alpha);
    atomAddF(accd + c4 * 4 + 3, v.w * alpha);
  }
}

// ---------- node update: t = h + elu(acc + bias); reduce sum/sumsq ----------
// 4 elements per thread (vectorized b128), block = 256 threads.
__global__ __launch_bounds__(256)
void node_update_kernel(const float* __restrict__ h, float* __restrict__ acc,
                        const float* __restrict__ bias, float* __restrict__ red,
                        int total4) {  // total/4
  __shared__ float ssum[256];
  __shared__ float ssq[256];
  const int idx = blockIdx.x * 256 + threadIdx.x;
  float lsum = 0.0f, lsq = 0.0f;
  if (idx < total4) {
    const int base = (idx * 4) & (DH - 1);
    const float4 a4 = ((const float4*)acc)[idx];
    const float4 h4 = ((const float4*)h)[idx];
    const float4 b4 = *(const float4*)(bias + base);
    float av[4] = {a4.x, a4.y, a4.z, a4.w};
    float hv[4] = {h4.x, h4.y, h4.z, h4.w};
    float bv[4] = {b4.x, b4.y, b4.z, b4.w};
    float tv[4];
#pragma unroll
    for (int j = 0; j < 4; ++j) {
      const float o  = av[j] + bv[j];
      const float hn = o > 0.0f ? o : (__expf(o) - 1.0f);  // elu
      const float t  = hv[j] + hn;                          // residual
      tv[j] = t;
      lsum += t;
      lsq  += t * t;
    }
    ((float4*)acc)[idx] = make_float4(tv[0], tv[1], tv[2], tv[3]);
  }
  ssum[threadIdx.x] = lsum;
  ssq[threadIdx.x]  = lsq;
  __syncthreads();
  for (int s = 128; s > 0; s >>= 1) {
    if (threadIdx.x < s) {
      ssum[threadIdx.x] += ssum[threadIdx.x + s];
      ssq[threadIdx.x]  += ssq[threadIdx.x + s];
    }
    __syncthreads();
  }
  if (threadIdx.x == 0) { atomAddF(&red[0], ssum[0]); atomAddF(&red[1], ssq[0]); }
}

// ---------- graph norm apply: h = (t - mu) * rstd * g + beta ----------
__global__ __launch_bounds__(256)
void norm_apply_kernel(const float* __restrict__ t, float* __restrict__ h,
                       const float* __restrict__ g, const float* __restrict__ beta,
                       const float* __restrict__ red, int total4) {
  const int idx = blockIdx.x * 256 + threadIdx.x;
  if (idx >= total4) return;
  const float inv_n = 1.0f / (float)(total4 * 4);
  const float mu    = red[0] * inv_n;
  const float var   = red[1] * inv_n - mu * mu;
  const float rstd  = rsqrtf(var + 1e-5f);
  const int base = (idx * 4) & (DH - 1);
  const float4 t4 = ((const float4*)t)[idx];
  const float4 g4 = *(const float4*)(g + base);
  const float4 b4 = *(const float4*)(beta + base);
  float4 o4;
  o4.x = (t4.x - mu) * rstd * g4.x + b4.x;
  o4.y = (t4.y - mu) * rstd * g4.y + b4.y;
  o4.z = (t4.z - mu) * rstd * g4.z + b4.z;
  o4.w = (t4.w - mu) * rstd * g4.w + b4.w;
  ((float4*)h)[idx] = o4;
}

// ---------- host ----------
extern "C" void kernel_launch(void* const* d_in, const int* in_sizes, int n_in,
                              void* d_out, int out_size, void* d_ws, size_t ws_size,
                              hipStream_t stream) {
  // setup_inputs() dict order:
  const float* x     = (const float*)d_in[0];
  const int*   eidx  = (const int*)d_in[1];
  const float* ea    = (const float*)d_in[2];
  const float* W_in  = (const float*)d_in[3];
  const float* b_in  = (const float*)d_in[4];
  const float* W_out = (const float*)d_in[5];
  const float* b_out = (const float*)d_in[6];
  struct Layer { const float *Wl, *bl, *Wr, *br, *We, *att, *bias, *g, *beta; } L[2];
  for (int i = 0; i < 2; ++i) {
    const int o = 7 + 9 * i;
    L[i].Wl   = (const float*)d_in[o + 0];
    L[i].bl   = (const float*)d_in[o + 1];
    L[i].Wr   = (const float*)d_in[o + 2];
    L[i].br   = (const float*)d_in[o + 3];
    L[i].We   = (const float*)d_in[o + 4];
    L[i].att  = (const float*)d_in[o + 5];
    L[i].bias = (const float*)d_in[o + 6];
    L[i].g    = (const float*)d_in[o + 7];
    L[i].beta = (const float*)d_in[o + 8];
  }
  const int* srcs = eidx;
  const int* dsts = eidx + EE;

  // workspace layout (floats), total ~221 MB
  const size_t NM = (size_t)NN * DH;     // 12.8M
  float*    h      = (float*)d_ws;
  float*    acc    = h + NM;
  float*    xl     = acc + NM;
  float*    xr     = xl + NM;
  float*    logits = xr + NM;            // Etot*NH, reused as p
  unsigned* mx     = (unsigned*)(logits + (size_t)ETOT * NH);
  float*    den    = (float*)(mx + (size_t)NN * NH);
  float*    easum  = den + (size_t)NN * NH;
  float*    red    = easum + ED;

  // self-loop edge_attr mean
  hipMemsetAsync(easum, 0, ED * sizeof(float), stream);
  ea_sum_kernel<<<1024, 256, 0, stream>>>(ea, easum);
  ea_scale_kernel<<<1, 32, 0, stream>>>(easum);

  // h = relu(x @ W_in + b_in)
  dim3 gg((NN + 127) / 128, DH / 64);
  gemm_bias_act<1><<<gg, 256, 0, stream>>>(x, W_in, b_in, h, NN, DIN, DH);

  const int etot_threads = ETOT * NH;
  const int eblocks = (etot_threads + 255) / 256;
  const int total4  = (int)(NM / 4);
  const int nblocks = (total4 + 255) / 256;

  for (int li = 0; li < 2; ++li) {
    const Layer& l = L[li];
    gemm_bias_act<0><<<gg, 256, 0, stream>>>(h, l.Wl, l.bl, xl, NN, DH, DH);
    gemm_bias_act<0><<<gg, 256, 0, stream>>>(h, l.Wr, l.br, xr, NN, DH, DH);

    hipMemsetAsync(mx, 0, (size_t)NN * NH * sizeof(unsigned), stream);  // fenc min
    hipMemsetAsync(den, 0, (size_t)NN * NH * sizeof(float), stream);
    hipMemsetAsync(acc, 0, NM * sizeof(float), stream);
    hipMemsetAsync(red, 0, 2 * sizeof(float), stream);

    edge_logits_kernel<<<eblocks, 256, 0, stream>>>(srcs, dsts, ea, easum, xl, xr,
                                                    l.We, l.att, logits, mx);
    edge_p_kernel<<<eblocks, 256, 0, stream>>>(dsts, logits, mx, den);
    edge_agg_kernel<<<eblocks, 256, 0, stream>>>(srcs, dsts, logits, den, xl, acc);

    node_update_kernel<<<nblocks, 256, 0, stream>>>(h, acc, l.bias, red, total4);
    norm_apply_kernel<<<nblocks, 256, 0, stream>>>(acc, h, l.g, l.beta, red, total4);
  }

  // out = h @ W_out + b_out
  gemm_bias_act<0><<<gg, 256, 0, stream>>>(h, W_out, b_out, (float*)d_out, NN, DH, DH);
}